// YOLOLoss_82592221102671
// MI455X (gfx1250) — compile-verified
//
#include <hip/hip_runtime.h>
#include <hip/hip_bf16.h>

// ---------------- CDNA5 wave32 reduction via V_WMMA_F32_16X16X4_F32 ----------
typedef __attribute__((ext_vector_type(2))) float v2f;
typedef __attribute__((ext_vector_type(8))) float v8f;

__device__ __forceinline__ float wave_reduce_sum(float v) {
#if __has_builtin(__builtin_amdgcn_wmma_f32_16x16x4_f32)
    // A (16x4 f32): lane L, VGPR0 -> K = (L<16 ? 0 : 2), M = L%16.  VGPR1 = 0.
    // B (4x16) = all ones  =>  D[m][n] = v_m + v_{m+16}  (same in every column)
    v2f a; a.x = v;    a.y = 0.0f;
    v2f b; b.x = 1.0f; b.y = 1.0f;
    v8f c = {};
    c = __builtin_amdgcn_wmma_f32_16x16x4_f32(
        /*neg_a=*/false, a, /*neg_b=*/false, b,
        /*c_mod=*/(short)0, c, /*reuse_a=*/false, /*reuse_b=*/false);
    // lanes 0-15 hold D[0..7][lane]  = sum_{m=0..7}(v_m+v_{m+16})
    // lanes 16-31 hold D[8..15][.]   = sum_{m=8..15}(v_m+v_{m+16})
    float s = c[0] + c[1] + c[2] + c[3] + c[4] + c[5] + c[6] + c[7];
    s += __shfl_xor(s, 16, 32);
    return s;
#else
    for (int off = 16; off > 0; off >>= 1) v += __shfl_xor(v, off, 32);
    return v;
#endif
}

// ---------------- math helpers ------------------------------------------------
__device__ __forceinline__ float bce_logits(float x, float t) {
    return fmaxf(x, 0.0f) - x * t + log1pf(expf(-fabsf(x)));
}

__device__ __forceinline__ float ciou(float px, float py, float pw, float ph,
                                      float tx, float ty, float tw, float th) {
    float b1x1 = px - pw * 0.5f, b1x2 = px + pw * 0.5f;
    float b1y1 = py - ph * 0.5f, b1y2 = py + ph * 0.5f;
    float b2x1 = tx - tw * 0.5f, b2x2 = tx + tw * 0.5f;
    float b2y1 = ty - th * 0.5f, b2y2 = ty + th * 0.5f;
    float iw = fmaxf(fminf(b1x2, b2x2) - fmaxf(b1x1, b2x1), 0.0f);
    float ih = fmaxf(fminf(b1y2, b2y2) - fmaxf(b1y1, b2y1), 0.0f);
    float inter = iw * ih;
    float uni = pw * ph + tw * th - inter + 1e-16f;
    float iou = inter / uni;
    float cw = fmaxf(b1x2, b2x2) - fminf(b1x1, b2x1);
    float ch = fmaxf(b1y2, b2y2) - fminf(b1y1, b2y1);
    float c2 = cw * cw + ch * ch + 1e-16f;
    float dx = b2x1 + b2x2 - b1x1 - b1x2;
    float dy = b2y1 + b2y2 - b1y1 - b1y2;
    float rho2 = (dx * dx + dy * dy) * 0.25f;
    float da = atanf(tw / th) - atanf(pw / ph);
    float v = (4.0f / (3.14159f * 3.14159f)) * da * da;
    float alpha = v / (v - iou + (1.0f + 1e-16f));
    return iou - (rho2 / c2 + v * alpha);
}

// ---------------- kernels -----------------------------------------------------
__global__ void yolo_zero_kernel(float* __restrict__ p, int n) {
    int i = blockIdx.x * blockDim.x + threadIdx.x;
    if (i < n) p[i] = 0.0f;
}

// One thread per candidate (5 offsets x 3 anchors x nt targets) for one scale.
__global__ void yolo_target_kernel(const float* __restrict__ pred,
                                   const float* __restrict__ targets,
                                   const float* __restrict__ anchors,
                                   float* __restrict__ scalars,  // [0..2]=lbox,[3..5]=nval,[6..8]=lcls
                                   float* __restrict__ tobj,     // (B,3,H,W) for this scale
                                   int nt, int Wg, int Hg, float stride,
                                   int anc_base, int scale_idx) {
    const int idx = blockIdx.x * blockDim.x + threadIdx.x;
    const int total = 5 * 3 * nt;
    float lbox_c = 0.0f, nval_c = 0.0f, lcls_c = 0.0f;
    if (idx < total) {
        const int o = idx / (3 * nt);
        const int rem = idx % (3 * nt);
        const int a = rem / nt;
        const int t = rem % nt;
        const float* tg = targets + t * 6;
        const float tb = tg[0], tc = tg[1];
        const float gx = tg[2] * (float)Wg, gy = tg[3] * (float)Hg;
        const float gw = tg[4] * (float)Wg, gh = tg[5] * (float)Hg;
        const float ax = anchors[(anc_base + a) * 2 + 0] / stride;
        const float ay = anchors[(anc_base + a) * 2 + 1] / stride;
        const float rx = gw / ax, ry = gh / ay;
        const float rmax = fmaxf(fmaxf(rx, 1.0f / rx), fmaxf(ry, 1.0f / ry));
        const bool amask = rmax < 4.0f;
        const float gxi = (float)Wg - gx, gyi = (float)Hg - gy;
        bool sel;
        switch (o) {
            case 0:  sel = true; break;
            case 1:  sel = (fmodf(gx, 1.0f) < 0.5f) && (gx > 1.0f);  break;
            case 2:  sel = (fmodf(gy, 1.0f) < 0.5f) && (gy > 1.0f);  break;
            case 3:  sel = (fmodf(gxi, 1.0f) < 0.5f) && (gxi > 1.0f); break;
            default: sel = (fmodf(gyi, 1.0f) < 0.5f) && (gyi > 1.0f); break;
        }
        if (sel && amask) {
            const float offx_tab[5] = {0.f, 1.f, 0.f, -1.f, 0.f};
            const float offy_tab[5] = {0.f, 0.f, 1.f, 0.f, -1.f};
            const int gij_x = (int)(gx - offx_tab[o]);   // trunc, matches astype(int32)
            const int gij_y = (int)(gy - offy_tab[o]);
            const int gi = min(max(gij_x, 0), Wg - 1);
            const int gj = min(max(gij_y, 0), Hg - 1);
            const int b = (int)tb;
            const int ci = (int)tc;
            const int HW = Hg * Wg;
            const float* base = pred + (size_t)(b * 255 + a * 85) * (size_t)HW
                                     + (size_t)gj * Wg + gi;
            const float p0v = base[0];
            const float p1v = base[(size_t)HW];
            const float p2v = base[(size_t)2 * HW];
            const float p3v = base[(size_t)3 * HW];
            const float px = 1.0f / (1.0f + expf(-p0v));
            const float py = 1.0f / (1.0f + expf(-p1v));
            const float pw = expf(p2v) * ax;
            const float ph = expf(p3v) * ay;
            const float tx = gx - (float)gij_x;          // unclipped gij, like reference
            const float ty = gy - (float)gij_y;
            const float iou = ciou(px, py, pw, ph, tx, ty, gw, gh);
            lbox_c = 1.0f - iou;
            nval_c = 1.0f;
            __builtin_prefetch(base + (size_t)5 * HW, 0, 3);  // class logits row start
            float s = 0.0f;
            for (int c = 0; c < 80; ++c) {
                const float xl = base[(size_t)(5 + c) * HW];
                s += bce_logits(xl, (c == ci) ? 1.0f : 0.0f);
            }
            lcls_c = s;
            // tobj scatter-max (values >= 0 so uint compare == float compare)
            const float iou_d = fmaxf(iou, 0.0f);
            unsigned* addr = (unsigned*)&tobj[((size_t)(b * 3 + a) * Hg + gj) * Wg + gi];
            atomicMax(addr, __float_as_uint(iou_d));
        }
    }
    // whole wave active here (no early returns) -> EXEC all-ones for WMMA
    const float r0 = wave_reduce_sum(lbox_c);
    const float r1 = wave_reduce_sum(nval_c);
    const float r2 = wave_reduce_sum(lcls_c);
    if ((threadIdx.x & 31) == 0) {
        atomicAdd(&scalars[0 + scale_idx], r0);
        atomicAdd(&scalars[3 + scale_idx], r1);
        atomicAdd(&scalars[6 + scale_idx], r2);
    }
}

// Objectness BCE over all B*3*H*W cells of one scale; reads only channel 4 of 85.
__global__ void yolo_obj_kernel(const float* __restrict__ pred,
                                const float* __restrict__ tobj,
                                float* __restrict__ lobj_accum,
                                int Hg, int Wg) {
    const int HW = Hg * Wg;
    const int total = 16 * 3 * HW;
    const int n = blockIdx.x * blockDim.x + threadIdx.x;
    float v = 0.0f;
    if (n < total) {
        const int b = n / (3 * HW);
        const int rem = n % (3 * HW);
        const int a = rem / HW;
        const int p = rem % HW;
        const float xl = pred[(size_t)(b * 255 + a * 85 + 4) * (size_t)HW + p];
        v = bce_logits(xl, tobj[n]);
    }
    const float r = wave_reduce_sum(v);
    if ((threadIdx.x & 31) == 0) atomicAdd(lobj_accum, r);
}

__global__ void yolo_finalize_kernel(const float* __restrict__ scalars,
                                     float* __restrict__ out) {
    const float counts[3] = {16.0f * 3 * 6400.0f, 16.0f * 3 * 1600.0f, 16.0f * 3 * 400.0f};
    float lbox = 0.0f, lobj = 0.0f, lcls = 0.0f;
    for (int s = 0; s < 3; ++s) {
        const float nv = scalars[3 + s];
        lbox += scalars[0 + s] / fmaxf(nv, 1.0f);
        lcls += scalars[6 + s] / fmaxf(nv * 80.0f, 1.0f);
        lobj += scalars[9 + s] / counts[s];
    }
    lbox *= 0.05f;
    lcls *= 0.5f;
    out[0] = lbox + lobj + lcls;
    out[1] = lbox;
    out[2] = lobj;
    out[3] = lcls;
}

// ---------------- host launch -------------------------------------------------
extern "C" void kernel_launch(void* const* d_in, const int* in_sizes, int n_in,
                              void* d_out, int out_size, void* d_ws, size_t ws_size,
                              hipStream_t stream) {
    const float* preds[3] = {(const float*)d_in[0], (const float*)d_in[1], (const float*)d_in[2]};
    const float* targets = (const float*)d_in[3];
    const float* anchors = (const float*)d_in[4];
    const int nt = in_sizes[3] / 6;   // targets: (N, 6)

    float* ws = (float*)d_ws;
    float* scalars = ws;              // 16 floats reserved for accumulators
    const int Wg[3] = {80, 40, 20};
    const float strf[3] = {8.0f, 16.0f, 32.0f};
    float* tobj[3];
    size_t off = 16;
    int counts[3];
    for (int s = 0; s < 3; ++s) {
        counts[s] = 16 * 3 * Wg[s] * Wg[s];
        tobj[s] = ws + off;
        off += (size_t)counts[s];
    }
    const int ztotal = (int)off;      // 403,216 floats ~1.6 MB

    yolo_zero_kernel<<<(ztotal + 255) / 256, 256, 0, stream>>>(ws, ztotal);

    const int cand = 5 * 3 * nt;      // 4500 per scale
    for (int s = 0; s < 3; ++s) {
        yolo_target_kernel<<<(cand + 255) / 256, 256, 0, stream>>>(
            preds[s], targets, anchors, scalars, tobj[s],
            nt, Wg[s], Wg[s], strf[s], s * 3, s);
    }
    for (int s = 0; s < 3; ++s) {
        yolo_obj_kernel<<<(counts[s] + 255) / 256, 256, 0, stream>>>(
            preds[s], tobj[s], &scalars[9 + s], Wg[s], Wg[s]);
    }
    yolo_finalize_kernel<<<1, 1, 0, stream>>>(scalars, (float*)d_out);
}